// DeepHamCritic_66400194396843
// MI455X (gfx1250) — compile-verified
//
#include <hip/hip_runtime.h>
#include <hip/hip_bf16.h>
#include <math.h>

// ---------------- problem constants (from reference) ----------------
#define NN   1000      // nodes
#define FIN  128       // input features
#define DD   512       // GCN hidden dim
#define HH   256       // dense hidden dim
#define ALPHA_NS 0.1f  // LeakyReLU slope

typedef float v2f __attribute__((ext_vector_type(2)));
typedef float v8f __attribute__((ext_vector_type(8)));

// ---------------- CSR build (once, reused by all 3 layers) ----------------
__global__ void k_zero_cnt(int* __restrict__ cnt, int n) {
    int i = blockIdx.x * blockDim.x + threadIdx.x;
    if (i < n) cnt[i] = 0;
}

__global__ void k_count_int(const long long* __restrict__ dst,
                            int* __restrict__ cnt, int E) {
    int e = blockIdx.x * blockDim.x + threadIdx.x;
    if (e < E) atomicAdd(&cnt[(int)dst[e]], 1);
}

// dinv[i] = rsqrt(deg_i), deg_i = in-degree + 1 (self loop)
__global__ void k_make_dinv(const int* __restrict__ cnt,
                            float* __restrict__ dinv, int n) {
    int i = blockIdx.x * blockDim.x + threadIdx.x;
    if (i < n) dinv[i] = __frsqrt_rn((float)(cnt[i] + 1));
}

// single-block 1024-wide Hillis-Steele scan -> rowptr (exclusive) + cursor copy
__global__ __launch_bounds__(1024) void k_scan(const int* __restrict__ cnt,
                                               int* __restrict__ rowptr,
                                               int* __restrict__ cursor, int n) {
    __shared__ int s[1024];
    const int t = threadIdx.x;
    s[t] = (t < n) ? cnt[t] : 0;
    __syncthreads();
    for (int off = 1; off < 1024; off <<= 1) {
        int v = (t >= off) ? s[t - off] : 0;
        __syncthreads();
        s[t] += v;
        __syncthreads();
    }
    if (t == 0) rowptr[0] = 0;
    if (t < n) {
        rowptr[t + 1] = s[t];                 // inclusive -> rowptr[t+1]
        cursor[t] = (t == 0) ? 0 : s[t - 1];  // exclusive start
    }
}

// bucket-fill edges by destination; precompute per-edge norm
__global__ void k_fill(const long long* __restrict__ src,
                       const long long* __restrict__ dst,
                       const float* __restrict__ dinv,
                       int* __restrict__ cursor,
                       int* __restrict__ ssrc, float* __restrict__ snorm, int E) {
    int e = blockIdx.x * blockDim.x + threadIdx.x;
    if (e >= E) return;
    const int s = (int)src[e];
    const int d = (int)dst[e];
    const int pos = atomicAdd(&cursor[d], 1);
    ssrc[pos]  = s;
    snorm[pos] = dinv[s] * dinv[d];
}

// ---------------- fp32 WMMA GEMM:  C[M,N] = A[M,K] @ B[K,N] ----------------
// one wave per 16x16 C tile; K steps of 4 via V_WMMA_F32_16X16X4_F32.
// A,B (<= 3 MB) stay L2-resident, so redundant tile re-reads are on-chip.
__global__ __launch_bounds__(32) void k_gemm_wmma_f32(
    const float* __restrict__ A, const float* __restrict__ B,
    float* __restrict__ C, int M, int K, int N)
{
    const int lane = threadIdx.x;          // 0..31
    const int half = lane >> 4;            // 0: K {0,1}; 1: K {2,3}
    const int tm   = lane & 15;
    const int m0   = blockIdx.y * 16;
    const int n0   = blockIdx.x * 16;

    int arow = m0 + tm;
    if (arow >= M) arow = M - 1;           // clamp; stores guarded below
    const float* __restrict__ Arow = A + (size_t)arow * K;

    v8f acc = {};
    for (int k0 = 0; k0 < K; k0 += 4) {
        const int ka = k0 + 2 * half;
        v2f a;
        a.x = Arow[ka];
        a.y = Arow[ka + 1];
        v2f b;
        b.x = B[(size_t)ka       * N + n0 + tm];
        b.y = B[(size_t)(ka + 1) * N + n0 + tm];
        acc = __builtin_amdgcn_wmma_f32_16x16x4_f32(
            false, a, false, b, (short)0, acc, false, false);
    }

#pragma unroll
    for (int v = 0; v < 8; ++v) {
        const int row = m0 + v + 8 * half;
        if (row < M) C[(size_t)row * N + n0 + tm] = acc[v];
    }
}

// ---------------- fused gather + self-loop + bias + tanh ----------------
// out[n,f] = tanh( dinv[n]^2 * h[n,f] + sum_e norm[e]*h[src[e],f] + b[f] )
// one block per node, 128 threads x float4 = 512 features.  No atomics;
// h (2 MB) is L2-resident so the random src-row reads stay on-chip.
__global__ __launch_bounds__(128) void k_gather_tanh(
    const int* __restrict__ rowptr, const int* __restrict__ ssrc,
    const float* __restrict__ snorm, const float* __restrict__ dinv,
    const float* __restrict__ h, const float* __restrict__ bias,
    float* __restrict__ out)
{
    const int node = blockIdx.x;
    const int c4   = threadIdx.x;          // 0..127 -> feats 4*c4..4*c4+3
    const float di2 = dinv[node] * dinv[node];

    float4 acc = *((const float4*)(h + (size_t)node * DD) + c4);
    acc.x *= di2; acc.y *= di2; acc.z *= di2; acc.w *= di2;

    const int beg = rowptr[node];
    const int end = rowptr[node + 1];
    for (int e = beg; e < end; ++e) {
        const int   s = ssrc[e];           // uniform across wave -> broadcast
        const float w = snorm[e];
        const float4 hv = *((const float4*)(h + (size_t)s * DD) + c4);
        acc.x += w * hv.x; acc.y += w * hv.y;
        acc.z += w * hv.z; acc.w += w * hv.w;
    }

    const float4 b4 = ((const float4*)bias)[c4];
    float4 o;
    o.x = tanhf(acc.x + b4.x);
    o.y = tanhf(acc.y + b4.y);
    o.z = tanhf(acc.z + b4.z);
    o.w = tanhf(acc.w + b4.w);
    ((float4*)(out + (size_t)node * DD))[c4] = o;
}

// ---------------- big dense GEMV: f[512000] @ Wd1[512000,256] ----------------
// 524 MB one-shot weight stream = whole-network bottleneck (~22.5 us at
// 23.3 TB/s).  Coalesced columns + non-temporal loads keep L2 clean for the
// GCN working set.
__global__ __launch_bounds__(256) void k_gemv_big(
    const float* __restrict__ f, const float* __restrict__ W,
    float* __restrict__ partial, int Ktotal, int kper)
{
    const int j = threadIdx.x;             // output column 0..255
    const int s = blockIdx.x;              // K slice
    const int kbeg = s * kper;
    int kend = kbeg + kper;
    if (kend > Ktotal) kend = Ktotal;
    const float* __restrict__ wp = W + (size_t)kbeg * HH + j;
    float acc = 0.0f;
    for (int k = kbeg; k < kend; ++k) {
        acc += f[k] * __builtin_nontemporal_load(wp);
        wp += HH;
    }
    partial[(size_t)s * HH + j] = acc;
}

__global__ __launch_bounds__(256) void k_reduce_lrelu(
    const float* __restrict__ partial, const float* __restrict__ bias,
    float* __restrict__ f1, int S)
{
    const int j = threadIdx.x;
    float acc = bias[j];
    for (int s = 0; s < S; ++s) acc += partial[(size_t)s * HH + j];
    f1[j] = acc > 0.0f ? acc : ALPHA_NS * acc;
}

// ---------------- fused small MLP head (single workgroup, 256 threads) -------
__global__ __launch_bounds__(256) void k_mlp_head(
    const float* __restrict__ f1,
    const float* __restrict__ Wd2, const float* __restrict__ bd2,
    const float* __restrict__ Wd3, const float* __restrict__ bd3,
    const float* __restrict__ Wd4, const float* __restrict__ bd4,
    float* __restrict__ out)
{
    __shared__ float sf[HH];
    __shared__ float sg[HH];
    const int j = threadIdx.x;

    sf[j] = f1[j];
    __syncthreads();

    float acc = bd2[j];
    for (int k = 0; k < HH; ++k) acc += sf[k] * Wd2[(size_t)k * HH + j];
    acc = acc > 0.0f ? acc : ALPHA_NS * acc;
    sg[j] = acc;
    __syncthreads();

    acc = bd3[j];
    for (int k = 0; k < HH; ++k) acc += sg[k] * Wd3[(size_t)k * HH + j];
    acc = acc > 0.0f ? acc : ALPHA_NS * acc;
    sf[j] = acc * Wd4[j];                  // elementwise product for final dot
    __syncthreads();

    if (j == 0) {
        float s = bd4[0];
        for (int k = 0; k < HH; ++k) s += sf[k];
        out[0] = s;
    }
}

// ---------------- driver ----------------
extern "C" void kernel_launch(void* const* d_in, const int* in_sizes, int n_in,
                              void* d_out, int out_size, void* d_ws, size_t ws_size,
                              hipStream_t stream) {
    const float*     x    = (const float*)d_in[0];
    const long long* ei   = (const long long*)d_in[1];
    const float*     W1   = (const float*)d_in[2];
    const float*     b1   = (const float*)d_in[3];
    const float*     W2   = (const float*)d_in[4];
    const float*     b2   = (const float*)d_in[5];
    const float*     W3   = (const float*)d_in[6];
    const float*     b3   = (const float*)d_in[7];
    const float*     Wd1  = (const float*)d_in[8];
    const float*     bd1  = (const float*)d_in[9];
    const float*     Wd2  = (const float*)d_in[10];
    const float*     bd2  = (const float*)d_in[11];
    const float*     Wd3  = (const float*)d_in[12];
    const float*     bd3  = (const float*)d_in[13];
    const float*     Wd4  = (const float*)d_in[14];
    const float*     bd4  = (const float*)d_in[15];
    float*           out  = (float*)d_out;

    const int E = in_sizes[1] / 2;
    const long long* esrc = ei;
    const long long* edst = ei + E;

    // workspace layout (4-byte word offsets)
    float* ws = (float*)d_ws;
    float* dinv    = ws;                                   // [1024]
    int*   cnt     = (int*)(ws + 1024);                    // [1024]
    int*   rowptr  = (int*)(ws + 2048);                    // [1024] (need 1001)
    int*   cursor  = (int*)(ws + 3072);                    // [1024]
    int*   ssrc    = (int*)(ws + 4096);                    // [E]
    float* snorm   = ws + 4096 + ((E + 255) & ~255);       // [E]
    float* hbuf    = snorm + ((E + 255) & ~255);           // [NN*DD]
    float* hcur    = hbuf + (size_t)NN * DD;               // [NN*DD]
    float* partial = hcur + (size_t)NN * DD;               // [512*HH]
    float* f1      = partial + 512 * HH;                   // [HH]

    const int T = 256;

    // --- CSR build + normalization (once) ---
    k_zero_cnt <<<(NN + T - 1) / T, T, 0, stream>>>(cnt, NN);
    k_count_int<<<(E  + T - 1) / T, T, 0, stream>>>(edst, cnt, E);
    k_make_dinv<<<(NN + T - 1) / T, T, 0, stream>>>(cnt, dinv, NN);
    k_scan     <<<1, 1024, 0, stream>>>(cnt, rowptr, cursor, NN);
    k_fill     <<<(E + T - 1) / T, T, 0, stream>>>(esrc, edst, dinv, cursor,
                                                   ssrc, snorm, E);

    const dim3 gemmGrid(DD / 16, (NN + 15) / 16);   // 32 x 63 WMMA tiles

    // --- layer 1 ---
    k_gemm_wmma_f32<<<gemmGrid, 32, 0, stream>>>(x, W1, hbuf, NN, FIN, DD);
    k_gather_tanh  <<<NN, 128, 0, stream>>>(rowptr, ssrc, snorm, dinv,
                                            hbuf, b1, hcur);
    // --- layer 2 ---
    k_gemm_wmma_f32<<<gemmGrid, 32, 0, stream>>>(hcur, W2, hbuf, NN, DD, DD);
    k_gather_tanh  <<<NN, 128, 0, stream>>>(rowptr, ssrc, snorm, dinv,
                                            hbuf, b2, hcur);
    // --- layer 3 ---
    k_gemm_wmma_f32<<<gemmGrid, 32, 0, stream>>>(hcur, W3, hbuf, NN, DD, DD);
    k_gather_tanh  <<<NN, 128, 0, stream>>>(rowptr, ssrc, snorm, dinv,
                                            hbuf, b3, hcur);

    // --- dense head ---
    const int Ktot = NN * DD;            // 512000 (hcur flattened row-major)
    const int S = 512;                   // K slices
    const int kper = (Ktot + S - 1) / S; // 1000
    k_gemv_big    <<<S, HH, 0, stream>>>(hcur, Wd1, partial, Ktot, kper);
    k_reduce_lrelu<<<1, HH, 0, stream>>>(partial, bd1, f1, S);
    k_mlp_head    <<<1, HH, 0, stream>>>(f1, Wd2, bd2, Wd3, bd3, Wd4, bd4, out);
}